// StructureEmbeddingLayer_44444321579161
// MI455X (gfx1250) — compile-verified
//
#include <hip/hip_runtime.h>

// ---------------------------------------------------------------------------
// StructureEmbeddingLayer for MI455X (gfx1250, wave32)
//   out (B=128, 64, 64, H=128) f32
//   - struct kernel: dense pairwise embedding via V_WMMA_F32_16X16X4_F32
//     (K=4 == NUM_STRUCT_FLOAT) + LDS-cached cate table + warp-LN
//   - virtual-edge kernel: row 0 / col 0
//   - bond kernel: per-edge embedding (two chained K=4 WMMAs for the 8->128
//     linear) + scatter-add via global f32 atomics
// Memory-bound problem (256 MB out @ 23.3 TB/s ~ 12us); WMMA keeps the
// linears off the VALU and the tables live in LDS (320 KB/WGP).
// ---------------------------------------------------------------------------

#define HDIM   128
#define BATCH  128
#define NNODE  63
#define NEDGE  128
#define MAXN   64
#define NN     (NNODE * NNODE)          // 3969 pairs per batch
#define TPB_S  ((NN + 15) / 16)         // 249 tiles of 16 positions per batch

typedef float v2f __attribute__((ext_vector_type(2)));
typedef float v8f __attribute__((ext_vector_type(8)));

// ---------------------------------------------------------------------------
// Dense pairwise structure embedding -> out[b, 1:, 1:, :]
// One wave handles a tile of 16 positions x all 128 channels (8 N-tiles).
// ---------------------------------------------------------------------------
__global__ __launch_bounds__(256, 1)
void struct_embed_kernel(const int* __restrict__ scate,      // (B,63,63,2)
                         const float* __restrict__ sfloat,   // (B,63,63,4)
                         const float* __restrict__ stable,   // (80,128)
                         const float* __restrict__ clw_g,
                         const float* __restrict__ clb_g,
                         const float* __restrict__ wlin_g,   // (128,4)
                         const float* __restrict__ blin_g,
                         const float* __restrict__ flw_g,
                         const float* __restrict__ flb_g,
                         float* __restrict__ out)            // (B,64,64,128)
{
    __shared__ float s_table[80 * HDIM];   // 40 KB categorical table
    __shared__ float s_par[5 * HDIM];      // clw, clb, flw, flb, lin bias
    __shared__ float s_w[HDIM * 4];        // linear weight (row-major HxK)

    for (int t = threadIdx.x; t < 80 * HDIM; t += 256) s_table[t] = stable[t];
    for (int t = threadIdx.x; t < HDIM * 4; t += 256)  s_w[t]     = wlin_g[t];
    if (threadIdx.x < HDIM) {
        s_par[threadIdx.x]            = clw_g[threadIdx.x];
        s_par[HDIM   + threadIdx.x]   = clb_g[threadIdx.x];
        s_par[2*HDIM + threadIdx.x]   = flw_g[threadIdx.x];
        s_par[3*HDIM + threadIdx.x]   = flb_g[threadIdx.x];
        s_par[4*HDIM + threadIdx.x]   = blin_g[threadIdx.x];
    }
    __syncthreads();

    const int wave = threadIdx.x >> 5;
    const int lane = threadIdx.x & 31;
    const int gtile = blockIdx.x * 8 + wave;
    if (gtile >= BATCH * TPB_S) return;            // uniform per wave

    const int b    = gtile / TPB_S;
    const int p0   = (gtile % TPB_S) * 16;
    const int mrow = lane & 15;                    // M (positions) / N (channels)
    const int half = lane >> 4;                    // selects K pair / M-half

    // --- stage per-position operands, one position per lane (m = lane&15) ---
    int  pm  = p0 + mrow;
    bool vm  = pm < NN;
    int  pmc = vm ? pm : (NN - 1);
    int  i   = pmc / NNODE, j = pmc % NNODE;

    const float* fptr = sfloat + ((size_t)b * NN + pmc) * 4 + 2 * half;
    __builtin_prefetch(sfloat + ((size_t)b * NN + pmc) * 4 + 64, 0, 0);
    v2f a;                                          // A tile: 16x4 f32
    a.x = vm ? fptr[0] : 0.0f;
    a.y = vm ? fptr[1] : 0.0f;

    const int* cptr = scate + ((size_t)b * NN + pmc) * 2;
    int my_i0  = cptr[0];                           // field0 rows [0,64)
    int my_i1  = 64 + cptr[1];                      // field1 rows [64,80)
    int my_row = ((b * MAXN + (i + 1)) * MAXN + (j + 1)) * HDIM;

    // --- redistribute metadata per accumulator row r (m = r + 8*half) ---
    int i0r[8], i1r[8], rowr[8]; bool vr[8];
#pragma unroll
    for (int r = 0; r < 8; ++r) {
        int src = r + 8 * half;
        i0r[r]  = __shfl(my_i0, src, 32);
        i1r[r]  = __shfl(my_i1, src, 32);
        rowr[r] = __shfl(my_row, src, 32);
        vr[r]   = (p0 + src) < NN;
    }

    // --- linear part: 8 channel tiles via V_WMMA_F32_16X16X4_F32 ---
    v8f lin[8];
#pragma unroll
    for (int c = 0; c < 8; ++c) {
        int ch = 16 * c + mrow;
        v2f bm;                                     // B tile: 4x16 f32
        bm.x = s_w[ch * 4 + 2 * half + 0];
        bm.y = s_w[ch * 4 + 2 * half + 1];
        v8f z = {0.f, 0.f, 0.f, 0.f, 0.f, 0.f, 0.f, 0.f};
        v8f acc = __builtin_amdgcn_wmma_f32_16x16x4_f32(
            false, a, false, bm, (short)0, z, false, false);
        float bias = s_par[4 * HDIM + ch];
#pragma unroll
        for (int r = 0; r < 8; ++r) acc[r] += bias;
        lin[c] = acc;
    }

    // --- pass 1: LN statistics over H=128 (8 tiles x 16-lane half-group) ---
    float sc[8], ssc[8], sl[8], ssl[8];
#pragma unroll
    for (int r = 0; r < 8; ++r) { sc[r] = ssc[r] = sl[r] = ssl[r] = 0.f; }
#pragma unroll
    for (int c = 0; c < 8; ++c) {
        int ch = 16 * c + mrow;
#pragma unroll
        for (int r = 0; r < 8; ++r) {
            float v = s_table[i0r[r] * HDIM + ch] + s_table[i1r[r] * HDIM + ch];
            sc[r] += v;  ssc[r] += v * v;
            float u = lin[c][r];
            sl[r] += u;  ssl[r] += u * u;
        }
    }
#pragma unroll
    for (int r = 0; r < 8; ++r) {
#pragma unroll
        for (int m = 8; m >= 1; m >>= 1) {          // stays inside 16-lane group
            sc[r]  += __shfl_xor(sc[r],  m, 32);
            ssc[r] += __shfl_xor(ssc[r], m, 32);
            sl[r]  += __shfl_xor(sl[r],  m, 32);
            ssl[r] += __shfl_xor(ssl[r], m, 32);
        }
    }
    float mc[8], rc[8], ml[8], rl[8];
    const float inv = 1.0f / HDIM;
#pragma unroll
    for (int r = 0; r < 8; ++r) {
        mc[r] = sc[r] * inv;
        rc[r] = rsqrtf(fmaxf(ssc[r] * inv - mc[r] * mc[r], 0.f) + 1e-5f);
        ml[r] = sl[r] * inv;
        rl[r] = rsqrtf(fmaxf(ssl[r] * inv - ml[r] * ml[r], 0.f) + 1e-5f);
    }

    // --- pass 2: normalize (re-gather from LDS) + coalesced store ---
#pragma unroll
    for (int c = 0; c < 8; ++c) {
        int ch = 16 * c + mrow;
        float clw = s_par[ch],            clb = s_par[HDIM + ch];
        float flw = s_par[2 * HDIM + ch], flb = s_par[3 * HDIM + ch];
#pragma unroll
        for (int r = 0; r < 8; ++r) {
            float v = s_table[i0r[r] * HDIM + ch] + s_table[i1r[r] * HDIM + ch];
            float o = (v - mc[r]) * rc[r] * clw + clb
                    + (lin[c][r] - ml[r]) * rl[r] * flw + flb;
            if (vr[r]) out[rowr[r] + ch] = o;
        }
    }
}

// ---------------------------------------------------------------------------
// Virtual edge rows: out[b,0,p,:] = out[b,p,0,:] = ve  (disjoint from struct)
// ---------------------------------------------------------------------------
__global__ void virtual_edge_kernel(const float* __restrict__ ve,
                                    float* __restrict__ out)
{
    int t = blockIdx.x * blockDim.x + threadIdx.x;
    if (t >= BATCH * MAXN * HDIM) return;
    int h   = t & (HDIM - 1);
    int pos = (t >> 7) & (MAXN - 1);
    int b   = t >> 13;
    float v = ve[h];
    out[((b * MAXN + 0)   * MAXN + pos) * HDIM + h] = v;
    out[((b * MAXN + pos) * MAXN + 0)   * HDIM + h] = v;
}

// ---------------------------------------------------------------------------
// Per-edge embedding + scatter-add.  8->128 linear = two chained K=4 WMMAs.
// 1024 tiles of 16 edges; duplicates accumulate via global_atomic_add_f32.
// ---------------------------------------------------------------------------
__global__ __launch_bounds__(256, 1)
void bond_embed_kernel(const int* __restrict__ bidx,        // (B,2,E)
                       const int* __restrict__ bcate,       // (B,E,3)
                       const float* __restrict__ bfloat,    // (B,E,8)
                       const float* __restrict__ bmask,     // (B,E)
                       const float* __restrict__ btable,    // (24,128)
                       const float* __restrict__ clw_g,
                       const float* __restrict__ clb_g,
                       const float* __restrict__ wlin_g,    // (128,8)
                       const float* __restrict__ blin_g,
                       const float* __restrict__ flw_g,
                       const float* __restrict__ flb_g,
                       float* __restrict__ out)
{
    __shared__ float s_table[24 * HDIM];   // 12 KB
    __shared__ float s_par[5 * HDIM];
    __shared__ float s_w[HDIM * 8];

    for (int t = threadIdx.x; t < 24 * HDIM; t += 256) s_table[t] = btable[t];
    for (int t = threadIdx.x; t < HDIM * 8; t += 256)  s_w[t]     = wlin_g[t];
    if (threadIdx.x < HDIM) {
        s_par[threadIdx.x]          = clw_g[threadIdx.x];
        s_par[HDIM   + threadIdx.x] = clb_g[threadIdx.x];
        s_par[2*HDIM + threadIdx.x] = flw_g[threadIdx.x];
        s_par[3*HDIM + threadIdx.x] = flb_g[threadIdx.x];
        s_par[4*HDIM + threadIdx.x] = blin_g[threadIdx.x];
    }
    __syncthreads();

    const int wave = threadIdx.x >> 5;
    const int lane = threadIdx.x & 31;
    const int gtile = blockIdx.x * 8 + wave;     // 1024 tiles exactly
    const int e0   = gtile * 16;                  // flattened b*E + e
    const int b    = e0 / NEDGE;                  // 16 | E -> no batch split
    const int mrow = lane & 15;
    const int half = lane >> 4;

    const int eg = e0 + mrow;
    const int e  = eg - b * NEDGE;

    const float* fptr = bfloat + (size_t)eg * 8;
    v2f a0, a1;                                   // 8 inputs = two K=4 chunks
    a0.x = fptr[2 * half + 0];
    a0.y = fptr[2 * half + 1];
    a1.x = fptr[4 + 2 * half + 0];
    a1.y = fptr[4 + 2 * half + 1];

    const int* cp = bcate + (size_t)eg * 3;
    int my_i0 = cp[0];                            // starts {0, 8, 16}
    int my_i1 = 8  + cp[1];
    int my_i2 = 16 + cp[2];
    int src_n = bidx[(b * 2 + 0) * NEDGE + e];
    int dst_n = bidx[(b * 2 + 1) * NEDGE + e];
    int my_row   = ((b * MAXN + src_n + 1) * MAXN + (dst_n + 1)) * HDIM;
    float my_msk = bmask[eg];

    int i0r[8], i1r[8], i2r[8], rowr[8]; float mkr[8];
#pragma unroll
    for (int r = 0; r < 8; ++r) {
        int src = r + 8 * half;
        i0r[r]  = __shfl(my_i0, src, 32);
        i1r[r]  = __shfl(my_i1, src, 32);
        i2r[r]  = __shfl(my_i2, src, 32);
        rowr[r] = __shfl(my_row, src, 32);
        mkr[r]  = __shfl(my_msk, src, 32);
    }

    v8f lin[8];
#pragma unroll
    for (int c = 0; c < 8; ++c) {
        int ch = 16 * c + mrow;
        v2f b0, b1;
        b0.x = s_w[ch * 8 + 2 * half + 0];
        b0.y = s_w[ch * 8 + 2 * half + 1];
        b1.x = s_w[ch * 8 + 4 + 2 * half + 0];
        b1.y = s_w[ch * 8 + 4 + 2 * half + 1];
        v8f acc = {0.f, 0.f, 0.f, 0.f, 0.f, 0.f, 0.f, 0.f};
        acc = __builtin_amdgcn_wmma_f32_16x16x4_f32(
            false, a0, false, b0, (short)0, acc, false, false);
        acc = __builtin_amdgcn_wmma_f32_16x16x4_f32(
            false, a1, false, b1, (short)0, acc, false, false);
        float bias = s_par[4 * HDIM + ch];
#pragma unroll
        for (int r = 0; r < 8; ++r) acc[r] += bias;
        lin[c] = acc;
    }

    float sc[8], ssc[8], sl[8], ssl[8];
#pragma unroll
    for (int r = 0; r < 8; ++r) { sc[r] = ssc[r] = sl[r] = ssl[r] = 0.f; }
#pragma unroll
    for (int c = 0; c < 8; ++c) {
        int ch = 16 * c + mrow;
#pragma unroll
        for (int r = 0; r < 8; ++r) {
            float v = s_table[i0r[r] * HDIM + ch]
                    + s_table[i1r[r] * HDIM + ch]
                    + s_table[i2r[r] * HDIM + ch];
            sc[r] += v;  ssc[r] += v * v;
            float u = lin[c][r];
            sl[r] += u;  ssl[r] += u * u;
        }
    }
#pragma unroll
    for (int r = 0; r < 8; ++r) {
#pragma unroll
        for (int m = 8; m >= 1; m >>= 1) {
            sc[r]  += __shfl_xor(sc[r],  m, 32);
            ssc[r] += __shfl_xor(ssc[r], m, 32);
            sl[r]  += __shfl_xor(sl[r],  m, 32);
            ssl[r] += __shfl_xor(ssl[r], m, 32);
        }
    }
    float mc[8], rc[8], ml[8], rl[8];
    const float inv = 1.0f / HDIM;
#pragma unroll
    for (int r = 0; r < 8; ++r) {
        mc[r] = sc[r] * inv;
        rc[r] = rsqrtf(fmaxf(ssc[r] * inv - mc[r] * mc[r], 0.f) + 1e-5f);
        ml[r] = sl[r] * inv;
        rl[r] = rsqrtf(fmaxf(ssl[r] * inv - ml[r] * ml[r], 0.f) + 1e-5f);
    }

#pragma unroll
    for (int c = 0; c < 8; ++c) {
        int ch = 16 * c + mrow;
        float clw = s_par[ch],            clb = s_par[HDIM + ch];
        float flw = s_par[2 * HDIM + ch], flb = s_par[3 * HDIM + ch];
#pragma unroll
        for (int r = 0; r < 8; ++r) {
            float v = s_table[i0r[r] * HDIM + ch]
                    + s_table[i1r[r] * HDIM + ch]
                    + s_table[i2r[r] * HDIM + ch];
            float o = ((v - mc[r]) * rc[r] * clw + clb
                     + (lin[c][r] - ml[r]) * rl[r] * flw + flb) * mkr[r];
            unsafeAtomicAdd(&out[rowr[r] + ch], o);   // global_atomic_add_f32
        }
    }
}

// ---------------------------------------------------------------------------
extern "C" void kernel_launch(void* const* d_in, const int* in_sizes, int n_in,
                              void* d_out, int out_size, void* d_ws, size_t ws_size,
                              hipStream_t stream) {
    const int*   bond_index      = (const int*)  d_in[0];
    const int*   bond_feat_cate  = (const int*)  d_in[1];
    const float* bond_feat_float = (const float*)d_in[2];
    const float* bond_mask       = (const float*)d_in[3];
    const int*   scate           = (const int*)  d_in[4];
    const float* sfloat          = (const float*)d_in[5];
    const float* bond_table      = (const float*)d_in[6];
    const float* bclw            = (const float*)d_in[7];
    const float* bclb            = (const float*)d_in[8];
    const float* blin_w          = (const float*)d_in[9];
    const float* blin_b          = (const float*)d_in[10];
    const float* bflw            = (const float*)d_in[11];
    const float* bflb            = (const float*)d_in[12];
    const float* ve              = (const float*)d_in[13];
    const float* stable          = (const float*)d_in[14];
    const float* sclw            = (const float*)d_in[15];
    const float* sclb            = (const float*)d_in[16];
    const float* slin_w          = (const float*)d_in[17];
    const float* slin_b          = (const float*)d_in[18];
    const float* sflw            = (const float*)d_in[19];
    const float* sflb            = (const float*)d_in[20];
    float* out = (float*)d_out;

    // 1) dense pairwise embedding (writes out[:,1:,1:,:])
    int sblocks = (BATCH * TPB_S + 7) / 8;              // 3984
    struct_embed_kernel<<<sblocks, 256, 0, stream>>>(
        scate, sfloat, stable, sclw, sclb, slin_w, slin_b, sflw, sflb, out);

    // 2) virtual edges (row 0 / col 0, disjoint region)
    int vthreads = BATCH * MAXN * HDIM;
    virtual_edge_kernel<<<(vthreads + 255) / 256, 256, 0, stream>>>(ve, out);

    // 3) edge scatter-add (stream-ordered after the dense write)
    bond_embed_kernel<<<128, 256, 0, stream>>>(
        bond_index, bond_feat_cate, bond_feat_float, bond_mask, bond_table,
        bclw, bclb, blin_w, blin_b, bflw, bflb, out);
}